// DifferentiableCircleRenderer_37082747634602
// MI455X (gfx1250) — compile-verified
//
#include <hip/hip_runtime.h>
#include <hip/hip_bf16.h>

// CDNA5 / gfx1250 circle renderer.
//
// Phase 1: dist^2 per (16-pixel x 16-circle) tile via V_WMMA_F32_16X16X4_F32:
//   A (16 pixels x K=4): row = [gx, gy, gx^2+gy^2, 1]
//   B (K=4 x 16 circles): col = [-2cx, -2cy, 1, cx^2+cy^2]
//   D[m][n] = (gx-cx)^2 + (gy-cy)^2   (exact f32; K=4 matches rank-2+bias)
// eff_mask = alpha * sigmoid((r - dist)/0.01)
//          = alpha * rcp(1 + exp2(dist*C - r*C)),  C = 100*log2(e)
// -> one med3 + sqrt + fma + exp2 + add + rcp + mul per element -> LDS.
//
// Phase 2: the composite canvas <- t*canvas + m*c is an affine map, so the
// order-dependent 128-circle scan is split associatively: waves 0-1 compose
// circles 0..63 per pixel, waves 2-3 compose circles 64..127 as (s, o_rgb),
// merged through LDS with one fma per channel. All 128 threads active,
// critical path = 64 sequential steps.

typedef __attribute__((ext_vector_type(2))) float v2f;
typedef __attribute__((ext_vector_type(8))) float v8f;

#define CANVAS      1024
#define NCIRC       128
#define PLANE       (CANVAS * CANVAS)
#define THREADS     128              // 4 waves (wave32)
#define PIX_PER_WG  64               // 4 waves x 16 pixels
#define LDS_STRIDE  133              // mod 64 = 5, coprime -> conflict-free
#define SIG_C       144.269504088896340736f   // 100 * log2(e)

__launch_bounds__(THREADS)
__global__ void circle_render_wmma_kernel(const float* __restrict__ centers, // (128,2)
                                          const float* __restrict__ radii,   // (128,)
                                          const float* __restrict__ colors,  // (128,4)
                                          float* __restrict__ out)           // (3,1024,1024)
{
    __shared__ float effLds[PIX_PER_WG * LDS_STRIDE];   // 34,048 B
    __shared__ float mergeLds[PIX_PER_WG][4];           // (s, oR, oG, oB) of upper half

    const int tid  = threadIdx.x;
    const int lane = tid & 31;
    const int wave = tid >> 5;
    const int l16  = lane & 15;
    const int hi   = lane >> 4;      // 0: lanes 0-15 (K=0,1) ; 1: lanes 16-31 (K=2,3)

    // ---------------- Phase 0: B fragments + per-circle scaled radius/alpha ----------
    v2f   bfrag[8];
    float rs[8], alp[8];
    #pragma unroll
    for (int b = 0; b < 8; ++b) {
        const int n  = b * 16 + l16;
        const float cx = centers[2 * n + 0];
        const float cy = centers[2 * n + 1];
        if (hi == 0) { bfrag[b].x = -2.0f * cx;  bfrag[b].y = -2.0f * cy;        }
        else         { bfrag[b].x = 1.0f;        bfrag[b].y = cx * cx + cy * cy; }
        rs[b]  = radii[n] * SIG_C;               // pre-scaled: arg = dist*C - rs
        alp[b] = colors[4 * n + 3];
    }

    // ---------------- Phase 1: WMMA dist^2 -> masks -> LDS ----------------
    const int wgPixBase = blockIdx.x * PIX_PER_WG;
    const int grpBase   = wave * 16;                 // this wave's 16 local pixels
    {
        const int   p  = wgPixBase + grpBase + l16;  // A row m = l16 -> this pixel
        const float gx = (float)(p & (CANVAS - 1)) * (1.0f / 1023.0f);
        const float gy = (float)(p >> 10)          * (1.0f / 1023.0f);
        v2f afrag;
        if (hi == 0) { afrag.x = gx;                afrag.y = gy;   }
        else         { afrag.x = gx * gx + gy * gy; afrag.y = 1.0f; }

        #pragma unroll
        for (int b = 0; b < 8; ++b) {
            v8f d2 = {0.0f, 0.0f, 0.0f, 0.0f, 0.0f, 0.0f, 0.0f, 0.0f};
            d2 = __builtin_amdgcn_wmma_f32_16x16x4_f32(
                     /*neg_a=*/false, afrag, /*neg_b=*/false, bfrag[b],
                     /*c_mod=*/(short)0, d2, /*reuse_a=*/false, /*reuse_b=*/false);

            const float rs_n = rs[b];
            const float a_n  = alp[b];
            // D layout: lane holds circle col = l16 (of block b),
            //           VGPR r holds pixel row m = r + 8*hi.
            #pragma unroll
            for (int r = 0; r < 8; ++r) {
                float dd   = __builtin_amdgcn_fmed3f(d2[r], 0.0f, 3.0e38f); // 1-op clamp
                float dist = __builtin_amdgcn_sqrtf(dd);
                float t    = fmaf(dist, SIG_C, -rs_n);          // (dist-r)*100*log2e
                float e    = __builtin_amdgcn_exp2f(t);         // exp(-(r-dist)/0.01)
                float m    = a_n * __builtin_amdgcn_rcpf(1.0f + e);
                const int pixLocal = grpBase + r + 8 * hi;
                effLds[pixLocal * LDS_STRIDE + b * 16 + l16] = m;
            }
        }
    }

    __syncthreads();

    // ---------------- Phase 2: associative split of the sequential composite --------
    const int half = tid >> 6;                   // waves 0-1: circles 0..63; 2-3: 64..127
    const int pix  = tid & 63;
    {
        const float* lrow = &effLds[pix * LDS_STRIDE + half * 64];
        const float* crow = colors + half * 64 * 4;   // wave-uniform -> scalar loads
        float s = 1.0f, oR = 0.0f, oG = 0.0f, oB = 0.0f;
        #pragma unroll 4
        for (int n = 0; n < 64; ++n) {           // in-order within the half
            const float m = lrow[n];             // ds_load_b32, conflict-free
            const float t = 1.0f - m;
            s *= t;
            oR = fmaf(oR, t, m * crow[4 * n + 0]);
            oG = fmaf(oG, t, m * crow[4 * n + 1]);
            oB = fmaf(oB, t, m * crow[4 * n + 2]);
        }
        if (half == 1) {                         // publish upper-half affine map
            mergeLds[pix][0] = s;
            mergeLds[pix][1] = oR;
            mergeLds[pix][2] = oG;
            mergeLds[pix][3] = oB;
        }
        __syncthreads();
        if (half == 0) {                         // apply halves in circle order
            const float s2  = mergeLds[pix][0];
            const float o2R = mergeLds[pix][1];
            const float o2G = mergeLds[pix][2];
            const float o2B = mergeLds[pix][3];
            const int p = wgPixBase + pix;
            out[0 * PLANE + p] = fmaf(s + oR, s2, o2R);   // canvas0 = 1
            out[1 * PLANE + p] = fmaf(s + oG, s2, o2G);
            out[2 * PLANE + p] = fmaf(s + oB, s2, o2B);
        }
    }
}

extern "C" void kernel_launch(void* const* d_in, const int* in_sizes, int n_in,
                              void* d_out, int out_size, void* d_ws, size_t ws_size,
                              hipStream_t stream) {
    (void)in_sizes; (void)n_in; (void)d_ws; (void)ws_size; (void)out_size;
    const float* centers = (const float*)d_in[0];   // (128,2) f32
    const float* radii   = (const float*)d_in[1];   // (128,)  f32
    const float* colors  = (const float*)d_in[2];   // (128,4) f32
    float*       out     = (float*)d_out;           // (3,1024,1024) f32

    const int blocks = PLANE / PIX_PER_WG;          // 16384
    circle_render_wmma_kernel<<<blocks, THREADS, 0, stream>>>(centers, radii, colors, out);
}